// DiffusionGraphConv_54657753809242
// MI455X (gfx1250) — compile-verified
//
#include <hip/hip_runtime.h>

typedef float v2f __attribute__((ext_vector_type(2)));
typedef float v8f __attribute__((ext_vector_type(8)));

// Problem constants (fixed by the reference)
constexpr int N_NODES = 20000;
constexpr int BATCH   = 16;
constexpr int FDIM    = 64;
constexpr int NNZ     = 640000;
constexpr int ODIM    = 64;
constexpr int NMAT    = 3;
constexpr int KDIM    = FDIM * NMAT;      // 192
constexpr int ROWS    = BATCH * N_NODES;  // 320000 output rows

// ---------------------------------------------------------------------------
// 1) CSR row_ptr from sorted sp_rows via per-row binary search
// ---------------------------------------------------------------------------
__global__ void rowptr_kernel(const int* __restrict__ rows, int* __restrict__ row_ptr) {
  int r = blockIdx.x * blockDim.x + threadIdx.x;
  if (r > N_NODES) return;
  int lo = 0, hi = NNZ;
  while (lo < hi) {
    int mid = (lo + hi) >> 1;
    if (rows[mid] < r) lo = mid + 1; else hi = mid;
  }
  row_ptr[r] = lo;                 // row_ptr[N] == NNZ (all rows < N)
}

// ---------------------------------------------------------------------------
// 2) SpMM in native [B, N, F] layout:
//    y[b,row,f] = alpha * sum_e vals[e] * x[b,cols[e],f]  (- sub[b,row,f])
//    256-thread block = 4 batch slices x 64 features; each wave sees a single
//    b, so the gather is scalar-base + f -> 256B coalesced segments, L2-
//    resident (82 MB/operand). cols/vals are wave-uniform -> scalar loads.
// ---------------------------------------------------------------------------
__global__ void spmm_kernel(const int* __restrict__ row_ptr, const int* __restrict__ cols,
                            const float* __restrict__ vals, const float* __restrict__ x,
                            const float* __restrict__ sub, float alpha,
                            float* __restrict__ y) {
  int row = blockIdx.x;
  int b   = blockIdx.y * 4 + (threadIdx.x >> 6);
  int f   = threadIdx.x & 63;
  long bbase = (long)b * N_NODES;
  long yi = (bbase + row) * FDIM + f;

  int e0 = row_ptr[row];
  int e1 = row_ptr[row + 1];
  float acc = 0.f;
  for (int e = e0; e < e1; ++e) {
    acc = fmaf(vals[e], x[(bbase + cols[e]) * FDIM + f], acc);
  }
  float r = alpha * acc;
  if (sub) r -= sub[yi];
  y[yi] = r;
}

// ---------------------------------------------------------------------------
// 3) out[r, o] = sum_c X[r, c] * W[c, o] + bias[o]
//    X[r, f*3+m] = xs_m[r*64 + f]  (never materialized; r = b*N+n directly).
//    One wave owns a 16x64 output tile: 4 v8f accumulators, 192 x
//    V_WMMA_F32_16X16X4_F32. Reduction regrouped by (m, f): each WMMA
//    consumes 4 contiguous columns f..f+3 of one diffusion matrix.
//
//    Fragment sourcing (zero per-step VALU):
//      A: one global_load_b64 per step — lane base = row*64 + 2*half,
//         f*4 bytes as the instruction immediate.
//      B: weights pre-swizzled in LDS into fragment pairs
//         sP[m][q][o] = { W[(2q)*3+m][o], W[(2q+1)*3+m][o] },  q = f/2 + half
//         -> one aligned ds_load_b64 per fragment, single base VGPR
//            (half*64 + l15 in v2f units) + compile-time immediate (< 48.5 KB).
//
//    ISA fragment layouts (CDNA5 ISA):
//      A 16x4 f32 : lane L holds row (L&15); VGPR0/1 = K {0,1} (lanes 0-15),
//                   K {2,3} (lanes 16-31).
//      B 4x16 f32 : mirrored; weight row of K=kk is (f+kk)*3 + m.
//      C/D        : VGPR v -> M=v (lanes 0-15) / M=v+8 (lanes 16-31), N=lane&15.
// ---------------------------------------------------------------------------
__global__ void __launch_bounds__(512)
gemm_wmma_kernel(const float* __restrict__ x0, const float* __restrict__ x1,
                 const float* __restrict__ x2, const float* __restrict__ W,
                 const float* __restrict__ bias, float* __restrict__ out) {
  __shared__ float sW[KDIM * ODIM];           // 48 KB of the 320 KB WGP LDS

  // Stage weights pre-swizzled into WMMA B-fragment order:
  // sW[((m*32 + q)*64 + o)*2 + e] = W[((2q+e)*3 + m)*64 + o]
  for (int i = threadIdx.x; i < KDIM * ODIM; i += 512) {
    int e = i & 1;
    int o = (i >> 1) & 63;
    int q = (i >> 7) & 31;
    int m = i >> 12;
    sW[i] = W[((2 * q + e) * NMAT + m) * ODIM + o];
  }
  __syncthreads();

  int wave = threadIdx.x >> 5;                // wave32: 16 waves/block
  int lane = threadIdx.x & 31;
  int gw   = blockIdx.x * 16 + wave;          // 20000 row tiles
  int r0   = gw << 4;

  int l15  = lane & 15;
  int half = lane >> 4;                       // 0: K={0,1}, 1: K={2,3}

  // Lane's A base: row (r0+l15), feature offset 2*half (8B aligned).
  long abase = (long)(r0 + l15) * FDIM + 2 * half;

  // Lane's B base in v2f units: q-half term (half*64) + output column l15.
  const v2f* sP = (const v2f*)sW;
  int pbase = half * 64 + l15;

  float bv0 = bias[ 0 + l15];
  float bv1 = bias[16 + l15];
  float bv2 = bias[32 + l15];
  float bv3 = bias[48 + l15];

  v8f acc0 = {}, acc1 = {}, acc2 = {}, acc3 = {};

#pragma unroll
  for (int m = 0; m < NMAT; ++m) {
    const float* xm = (m == 0) ? x0 : (m == 1) ? x1 : x2;  // literal after unroll
    const float* aptr = xm + abase;
#pragma unroll
    for (int f = 0; f < FDIM; f += 4) {
      v2f a = *(const v2f*)(aptr + f);        // columns f+2h, f+2h+1 of matrix m

      int i0 = pbase + m * 2048 + (f >> 1) * 64;   // v2f index; imm-foldable
      v2f b0 = sP[i0 +  0];
      acc0 = __builtin_amdgcn_wmma_f32_16x16x4_f32(false, a, false, b0,
                                                   (short)0, acc0, false, false);
      v2f b1 = sP[i0 + 16];
      acc1 = __builtin_amdgcn_wmma_f32_16x16x4_f32(false, a, false, b1,
                                                   (short)0, acc1, false, false);
      v2f b2 = sP[i0 + 32];
      acc2 = __builtin_amdgcn_wmma_f32_16x16x4_f32(false, a, false, b2,
                                                   (short)0, acc2, false, false);
      v2f b3 = sP[i0 + 48];
      acc3 = __builtin_amdgcn_wmma_f32_16x16x4_f32(false, a, false, b3,
                                                   (short)0, acc3, false, false);
    }
  }

#pragma unroll
  for (int v = 0; v < 8; ++v) {
    long row = r0 + v + half * 8;
    float* orow = out + row * ODIM;
    orow[ 0 + l15] = acc0[v] + bv0;
    orow[16 + l15] = acc1[v] + bv1;
    orow[32 + l15] = acc2[v] + bv2;
    orow[48 + l15] = acc3[v] + bv3;
  }
}

// ---------------------------------------------------------------------------
// Launch
// ---------------------------------------------------------------------------
extern "C" void kernel_launch(void* const* d_in, const int* in_sizes, int n_in,
                              void* d_out, int out_size, void* d_ws, size_t ws_size,
                              hipStream_t stream) {
  const float* inputs  = (const float*)d_in[0];   // [B,N,F] == x0 directly
  const int*   sp_rows = (const int*)d_in[1];
  const int*   sp_cols = (const int*)d_in[2];
  const float* sp_vals = (const float*)d_in[3];
  const float* weight  = (const float*)d_in[4];
  const float* biases  = (const float*)d_in[5];
  float* out = (float*)d_out;

  // Workspace layout: row_ptr | x1 | x2   (~164 MB)
  constexpr size_t X_BYTES = (size_t)ROWS * FDIM * sizeof(float); // 81.92 MB
  char* ws = (char*)d_ws;
  int*   row_ptr = (int*)ws;
  float* x1 = (float*)(ws + (1 << 17));
  float* x2 = (float*)(ws + (1 << 17) + X_BYTES);

  // 1) CSR row pointers
  rowptr_kernel<<<(N_NODES + 1 + 255) / 256, 256, 0, stream>>>(sp_rows, row_ptr);

  // 2) Chebyshev diffusion steps (x0 is the untouched input tensor)
  dim3 sgrid(N_NODES, BATCH / 4);
  spmm_kernel<<<sgrid, 256, 0, stream>>>(row_ptr, sp_cols, sp_vals, inputs, nullptr, 1.0f, x1);
  spmm_kernel<<<sgrid, 256, 0, stream>>>(row_ptr, sp_cols, sp_vals, x1,     inputs, 2.0f, x2);

  // 3) fused gather-GEMM with fp32 WMMA (one wave per 16x64 output tile)
  gemm_wmma_kernel<<<(ROWS / 16) / 16, 512, 0, stream>>>(inputs, x1, x2, weight, biases, out);
}